// DarcyGNN_24867860644039
// MI455X (gfx1250) — compile-verified
//
#include <hip/hip_runtime.h>
#include <cmath>

#define N_NODES 50000
#define N_EDGES 800000
#define HIDDEN  128
#define EDGE_DIM 4
#define N_LAYERS 4
#define MLP_HID 64

typedef __attribute__((ext_vector_type(2))) float v2f;
typedef __attribute__((ext_vector_type(8))) float v8f;

// h[n,j] = x[n]*embed_w[j] + embed_b[j]
__global__ void embed_kernel(const float* __restrict__ x,
                             const float* __restrict__ w,
                             const float* __restrict__ b,
                             float* __restrict__ h) {
  int i = blockIdx.x * blockDim.x + threadIdx.x;
  if (i >= N_NODES * HIDDEN) return;
  int n = i >> 7, j = i & 127;
  h[i] = fmaf(x[n], w[j], b[j]);
}

__global__ void zero_kernel(float* __restrict__ p, int n) {
  int i = blockIdx.x * blockDim.x + threadIdx.x;
  if (i < n) p[i] = 0.0f;
}

// Per-edge MLP: relu(ea @ W1 + b1) @ W2 + b2 -> softplus. Weights staged in LDS.
__global__ __launch_bounds__(256)
void edge_mlp_kernel(const float* __restrict__ edge_attr,
                     const float* __restrict__ ew1,  // [4][64]
                     const float* __restrict__ eb1,  // [64]
                     const float* __restrict__ ew2,  // [64]
                     const float* __restrict__ eb2,  // [1]
                     float* __restrict__ kout) {
  __shared__ float sw1[EDGE_DIM * MLP_HID];
  __shared__ float sb1[MLP_HID];
  __shared__ float sw2[MLP_HID];
  __shared__ float sb2;
  int t = threadIdx.x;
  for (int i = t; i < EDGE_DIM * MLP_HID; i += 256) sw1[i] = ew1[i];
  if (t < MLP_HID) { sb1[t] = eb1[t]; sw2[t] = ew2[t]; }
  if (t == 0) sb2 = eb2[0];
  __syncthreads();

  int e = blockIdx.x * 256 + t;
  if (e >= N_EDGES) return;
  const float4 a = ((const float4*)edge_attr)[e];  // 16B aligned, global_load_b128
  float acc = sb2;
  #pragma unroll 4
  for (int j = 0; j < MLP_HID; ++j) {
    float m = sb1[j];
    m = fmaf(a.x, sw1[0 * MLP_HID + j], m);
    m = fmaf(a.y, sw1[1 * MLP_HID + j], m);
    m = fmaf(a.z, sw1[2 * MLP_HID + j], m);
    m = fmaf(a.w, sw1[3 * MLP_HID + j], m);
    m = fmaxf(m, 0.0f);
    acc = fmaf(m, sw2[j], acc);
  }
  // stable softplus
  kout[e] = (acc > 20.0f) ? acc : log1pf(expf(acc));
}

// msg = k[e] * (h[src] - h[dst]); segment-sum via f32 atomics (L2-resident target).
__global__ __launch_bounds__(256)
void message_kernel(const float* __restrict__ hp,
                    float* __restrict__ hn,
                    const float* __restrict__ kv,
                    const long long* __restrict__ ei) {
  long long i = (long long)blockIdx.x * 256 + threadIdx.x;
  if (i >= (long long)N_EDGES * HIDDEN) return;
  int e = (int)(i >> 7);
  int c = (int)(i & 127);
  int src = (int)ei[e];
  int dst = (int)ei[N_EDGES + e];
  float m = kv[e] * (hp[src * HIDDEN + c] - hp[dst * HIDDEN + c]);
  atomicAdd(&hn[dst * HIDDEN + c], m);
}

// Output head: out = relu(h @ W1 + b1) @ W2 + b2 using V_WMMA_F32_16X16X4_F32.
// Block = 256 threads = 8 wave32 waves. Each block: 16 node rows (LDS-staged).
// Each wave: one 16-column tile of W1, 32 WMMA K-steps over K=128.
__global__ __launch_bounds__(256)
void head_kernel(const float* __restrict__ h,
                 const float* __restrict__ w1,   // [128][128]
                 const float* __restrict__ b1,   // [128]
                 const float* __restrict__ w2,   // [128]
                 const float* __restrict__ b2,   // [1]
                 float* __restrict__ out) {
  __shared__ float sh[16 * 132];   // 16 rows x 128, padded to 132 (bank spread)
  __shared__ float sout[16];
  const int node0 = blockIdx.x * 16;   // 50000 = 3125 * 16, no tail
  const int t = threadIdx.x;

  for (int i = t; i < 16 * 128; i += 256) {
    int r = i >> 7, c = i & 127;
    sh[r * 132 + c] = h[(node0 + r) * HIDDEN + c];
  }
  if (t < 16) sout[t] = 0.0f;
  __syncthreads();

  const int wave = t >> 5;           // 0..7 -> column tile
  const int lane = t & 31;
  const int half = lane >> 4;        // 0: K=0,1 / rows M=0..7 ; 1: K=2,3 / rows M=8..15
  const int lr   = lane & 15;
  const int col  = wave * 16 + lr;   // output column in W1 tile

  v8f acc = {};
  #pragma unroll 8
  for (int k0 = 0; k0 < HIDDEN; k0 += 4) {
    const int kk = k0 + 2 * half;
    v2f a, b;
    // A: 16x4 fp32 fragment (ISA 7.12.2: M across lanes, K in {VGPR, lane-half})
    a.x = sh[lr * 132 + kk];
    a.y = sh[lr * 132 + kk + 1];
    // B: 4x16 fp32 fragment (N across lanes, K in {VGPR, lane-half})
    b.x = w1[kk * HIDDEN + col];
    b.y = w1[(kk + 1) * HIDDEN + col];
    acc = __builtin_amdgcn_wmma_f32_16x16x4_f32(
        /*neg_a=*/false, a, /*neg_b=*/false, b,
        /*c_mod=*/(short)0, acc, /*reuse_a=*/false, /*reuse_b=*/false);
  }

  // Fused ReLU + (128 -> 1) projection: lane holds rows {half*8+g}, column `col`.
  const float bb = b1[col];
  const float ww = w2[col];
  #pragma unroll
  for (int g = 0; g < 8; ++g) {
    float v = fmaxf(acc[g] + bb, 0.0f);
    atomicAdd(&sout[g + 8 * half], v * ww);   // ds_add_f32
  }
  __syncthreads();
  if (t < 16) out[node0 + t] = sout[t] + b2[0];
}

extern "C" void kernel_launch(void* const* d_in, const int* in_sizes, int n_in,
                              void* d_out, int out_size, void* d_ws, size_t ws_size,
                              hipStream_t stream) {
  const float*     x       = (const float*)d_in[0];
  const long long* ei      = (const long long*)d_in[1];   // int64 [2, E]
  const float*     eattr   = (const float*)d_in[2];
  const float*     embed_w = (const float*)d_in[3];
  const float*     embed_b = (const float*)d_in[4];
  const float*     ew1     = (const float*)d_in[5];       // [L,4,64]
  const float*     eb1     = (const float*)d_in[6];       // [L,64]
  const float*     ew2     = (const float*)d_in[7];       // [L,64,1]
  const float*     eb2     = (const float*)d_in[8];       // [L,1]
  const float*     ow1     = (const float*)d_in[9];
  const float*     ob1     = (const float*)d_in[10];
  const float*     ow2     = (const float*)d_in[11];
  const float*     ob2     = (const float*)d_in[12];

  char* ws = (char*)d_ws;
  const size_t hbytes = (size_t)N_NODES * HIDDEN * sizeof(float);   // 25.6 MB
  float* hA   = (float*)ws;
  float* hB   = (float*)(ws + hbytes);
  float* kbuf = (float*)(ws + 2 * hbytes);                          // 3.2 MB

  const int NH = N_NODES * HIDDEN;
  embed_kernel<<<(NH + 255) / 256, 256, 0, stream>>>(x, embed_w, embed_b, hA);

  float* hp = hA;
  float* hn = hB;
  for (int l = 0; l < N_LAYERS; ++l) {
    edge_mlp_kernel<<<(N_EDGES + 255) / 256, 256, 0, stream>>>(
        eattr, ew1 + (size_t)l * EDGE_DIM * MLP_HID, eb1 + (size_t)l * MLP_HID,
        ew2 + (size_t)l * MLP_HID, eb2 + l, kbuf);
    zero_kernel<<<(NH + 255) / 256, 256, 0, stream>>>(hn, NH);
    const long long tot = (long long)N_EDGES * HIDDEN;
    message_kernel<<<(int)((tot + 255) / 256), 256, 0, stream>>>(hp, hn, kbuf, ei);
    float* tmp = hp; hp = hn; hn = tmp;
  }

  head_kernel<<<N_NODES / 16, 256, 0, stream>>>(hp, ow1, ob1, ow2, ob2, (float*)d_out);
}